// GNNLayer_41893111005363
// MI455X (gfx1250) — compile-verified
//
#include <hip/hip_runtime.h>

typedef __attribute__((ext_vector_type(2))) float v2f;
typedef __attribute__((ext_vector_type(8))) float v8f;

#define N_NODES 100000
#define N_EDGES 1600000
#define D       128
#define M_TILES (N_NODES / 16)   // 6250, exact

// ---------------------------------------------------------------------------
// 1) init: d_out <- 1.0 (absorbs the "+1.0" bias into the accumulator),
//          deg   <- 0.0
// ---------------------------------------------------------------------------
__global__ void gnn_init_kernel(float* __restrict__ out, float* __restrict__ deg) {
  int i = blockIdx.x * blockDim.x + threadIdx.x;
  if (i < N_NODES * D) out[i] = 1.0f;
  if (i < N_NODES)     deg[i] = 0.0f;
}

// ---------------------------------------------------------------------------
// 2) out-degree histogram over src (hardware f32 atomics, L2-resident)
// ---------------------------------------------------------------------------
__global__ void gnn_degree_kernel(const int* __restrict__ src,
                                  float* __restrict__ deg) {
  int e = blockIdx.x * blockDim.x + threadIdx.x;
  if (e < N_EDGES) {
    __hip_atomic_fetch_add(&deg[src[e]], 1.0f,
                           __ATOMIC_RELAXED, __HIP_MEMORY_SCOPE_AGENT);
  }
}

// ---------------------------------------------------------------------------
// 3) norm = rsqrt(max(deg, 1))   (in place)
// ---------------------------------------------------------------------------
__global__ void gnn_norm_kernel(float* __restrict__ deg) {
  int i = blockIdx.x * blockDim.x + threadIdx.x;
  if (i < N_NODES) deg[i] = rsqrtf(fmaxf(deg[i], 1.0f));
}

// ---------------------------------------------------------------------------
// 4) edge scatter: agg[dst] += x[src] * norm[src]
//    one wave32 per edge; each lane handles 4 consecutive floats (float4
//    gather, 4x global_atomic_add_f32 scatter). agg == d_out (pre-set to 1.0).
// ---------------------------------------------------------------------------
__global__ void gnn_scatter_kernel(const float4* __restrict__ x4,
                                   const int* __restrict__ src,
                                   const int* __restrict__ dst,
                                   const float* __restrict__ norm,
                                   float* __restrict__ agg) {
  int gid  = blockIdx.x * blockDim.x + threadIdx.x;
  int e    = gid >> 5;          // edge index
  int lane = gid & 31;          // 32 lanes * 4 floats = 128 features
  if (e >= N_EDGES) return;

  int s = src[e];
  int d = dst[e];
  float nv = norm[s];
  float4 v = x4[s * (D / 4) + lane];   // 16B gather, hits L2 (x is 51 MB)

  float* p = agg + (size_t)d * D + lane * 4;
  __hip_atomic_fetch_add(p + 0, v.x * nv, __ATOMIC_RELAXED, __HIP_MEMORY_SCOPE_AGENT);
  __hip_atomic_fetch_add(p + 1, v.y * nv, __ATOMIC_RELAXED, __HIP_MEMORY_SCOPE_AGENT);
  __hip_atomic_fetch_add(p + 2, v.z * nv, __ATOMIC_RELAXED, __HIP_MEMORY_SCOPE_AGENT);
  __hip_atomic_fetch_add(p + 3, v.w * nv, __ATOMIC_RELAXED, __HIP_MEMORY_SCOPE_AGENT);
}

// ---------------------------------------------------------------------------
// 5) GEMM: out = agg @ W^T + b, in place over d_out, full fp32 WMMA.
//    Each wave owns one 16-row M-tile and all 128 output columns:
//    8 accumulators (v8f), K-loop of 32 steps of V_WMMA_F32_16X16X4_F32.
//    W (64 KB) staged in LDS per workgroup (8 waves share it).
//
//    A layout (16x4 f32, ISA 7.12.2): lane l holds rows (l&15),
//      VGPR0/1 = K + (l>>4)*2, K + (l>>4)*2 + 1.
//    B layout (4x16): mirrored striping; B[k][n] = W[n][k] so a lane reads
//      the contiguous pair W[col*128 + k + (l>>4)*2 .. +1] -> ds_load_b64.
// ---------------------------------------------------------------------------
__global__ void __launch_bounds__(256)
gnn_gemm_kernel(const float* __restrict__ W,
                const float* __restrict__ bias,
                float* __restrict__ out /* agg in, result out */) {
  __shared__ float wlds[D * D];   // 64 KB of the WGP's 320 KB LDS

  // cooperative LDS fill: 4096 float4's across 256 threads
  {
    const float4* W4 = (const float4*)W;
    float4*       L4 = (float4*)wlds;
    for (int i = threadIdx.x; i < (D * D) / 4; i += 256) L4[i] = W4[i];
  }
  __syncthreads();

  const int wave = threadIdx.x >> 5;
  const int lane = threadIdx.x & 31;
  const int tile = blockIdx.x * 8 + wave;
  if (tile >= M_TILES) return;            // wave-uniform -> EXEC stays all-1s

  const int m0    = tile * 16;
  const int laneM = lane & 15;            // row within tile for A / col for B
  const int kofs  = (lane >> 4) * 2;      // K sub-offset per half-wave

  const float* arow = out + (size_t)(m0 + laneM) * D;  // this wave's agg rows

  v8f acc[8];
#pragma unroll
  for (int nt = 0; nt < 8; ++nt) acc[nt] = (v8f){0.f,0.f,0.f,0.f,0.f,0.f,0.f,0.f};

  for (int k = 0; k < D; k += 4) {
    v2f a = *(const v2f*)(arow + k + kofs);            // global_load_b64
#pragma unroll
    for (int nt = 0; nt < 8; ++nt) {
      const int col = nt * 16 + laneM;
      v2f b = *(const v2f*)(&wlds[col * D + k + kofs]); // ds_load_b64
      acc[nt] = __builtin_amdgcn_wmma_f32_16x16x4_f32(
          /*neg_a=*/false, a, /*neg_b=*/false, b,
          /*c_mod=*/(short)0, acc[nt],
          /*reuse_a=*/false, /*reuse_b=*/false);
    }
  }

  // C/D layout: VGPR r, lanes 0-15 -> M = m0+r, lanes 16-31 -> M = m0+8+r
  const int rowBase = m0 + (lane >> 4) * 8;
#pragma unroll
  for (int nt = 0; nt < 8; ++nt) {
    const int col = nt * 16 + laneM;
    const float bv = bias[col];
#pragma unroll
    for (int r = 0; r < 8; ++r) {
      out[(size_t)(rowBase + r) * D + col] = acc[nt][r] + bv;
    }
  }
}

// ---------------------------------------------------------------------------
// launch
// ---------------------------------------------------------------------------
extern "C" void kernel_launch(void* const* d_in, const int* in_sizes, int n_in,
                              void* d_out, int out_size, void* d_ws, size_t ws_size,
                              hipStream_t stream) {
  const float*  x    = (const float*)d_in[0];   // [100000,128]
  const int*    src  = (const int*)d_in[1];     // [1600000]
  const int*    dst  = (const int*)d_in[2];     // [1600000]
  const float*  W    = (const float*)d_in[3];   // [128,128]
  const float*  b    = (const float*)d_in[4];   // [128]
  float*        out  = (float*)d_out;           // [100000,128]
  float*        deg  = (float*)d_ws;            // 100000 floats (400 KB)

  (void)in_sizes; (void)n_in; (void)out_size; (void)ws_size;

  // 1) init out=1.0 (agg bias), deg=0
  {
    int n = N_NODES * D;
    gnn_init_kernel<<<(n + 255) / 256, 256, 0, stream>>>(out, deg);
  }
  // 2) degrees
  gnn_degree_kernel<<<(N_EDGES + 255) / 256, 256, 0, stream>>>(src, deg);
  // 3) norm
  gnn_norm_kernel<<<(N_NODES + 255) / 256, 256, 0, stream>>>(deg);
  // 4) scatter (one wave per edge)
  {
    long long threads = (long long)N_EDGES * 32;
    int blocks = (int)((threads + 255) / 256);
    gnn_scatter_kernel<<<blocks, 256, 0, stream>>>((const float4*)x, src, dst,
                                                   deg, out);
  }
  // 5) fp32 WMMA GEMM, in place on d_out
  {
    int blocks = (M_TILES + 7) / 8;   // 8 waves (tiles) per 256-thread block
    gnn_gemm_kernel<<<blocks, 256, 0, stream>>>(W, b, out);
  }
}